// SO3LocalActivation_47390669144374
// MI455X (gfx1250) — compile-verified
//
#include <hip/hip_runtime.h>
#include <cstdint>
#include <cstddef>
#include <utility>

// ---------------- problem constants (order = 3 fixed by reference) ----------
#define S_SPATIAL 13824      // 24*24*24
#define NBF       32         // B*F = 2*16
#define NIN       35         // _off(3)
#define NOUT      165        // _off(5)
#define THREADS   256

// ---------------- compile-time Clebsch-Gordan machinery ---------------------
constexpr double cfact(int n) { double r = 1.0; for (int k = 2; k <= n; ++k) r *= (double)k; return r; }

constexpr double csqrt_(double x) {
    if (x <= 0.0) return 0.0;
    double g = x < 1.0 ? 1.0 : x;
    for (int it = 0; it < 64; ++it) g = 0.5 * (g + x / g);
    return g;
}
constexpr int imax3(int a, int b, int c) { int m = a > b ? a : b; return m > c ? m : c; }
constexpr int imin3(int a, int b, int c) { int m = a < b ? a : b; return m < c ? m : c; }

// <j1 m1 j2 m2 | J M>  (Racah formula, matches reference _cg_val)
constexpr double cgval(int j1, int m1, int j2, int m2, int J, int M) {
    if (m1 + m2 != M) return 0.0;
    int dj = j1 > j2 ? j1 - j2 : j2 - j1;
    if (J < dj || J > j1 + j2) return 0.0;
    if (M < -J || M > J) return 0.0;
    double pref = (2.0 * J + 1.0) * cfact(J + j1 - j2) * cfact(J - j1 + j2) * cfact(j1 + j2 - J)
                  / cfact(j1 + j2 + J + 1);
    pref = csqrt_(pref);
    pref *= csqrt_(cfact(J + M) * cfact(J - M) * cfact(j1 - m1) * cfact(j1 + m1)
                   * cfact(j2 - m2) * cfact(j2 + m2));
    int kmin = imax3(0, j2 - J - m1, j1 - J + m2);
    int kmax = imin3(j1 + j2 - J, j1 - m1, j2 + m2);
    double s = 0.0;
    for (int k = kmin; k <= kmax; ++k) {
        double d = cfact(k) * cfact(j1 + j2 - J - k) * cfact(j1 - m1 - k)
                 * cfact(j2 + m2 - k) * cfact(J - j2 + m1 + k) * cfact(J - j1 - m2 + k);
        s += ((k & 1) ? -1.0 : 1.0) / d;
    }
    return pref * s;
}

struct CgAll { double v[3][3][5][5][5]; };   // [l1][l2][L][m1+2][m2+2]
constexpr CgAll make_cg() {
    CgAll t{};
    for (int l1 = 0; l1 < 3; ++l1)
        for (int l2 = 0; l2 < 3; ++l2)
            for (int L = 0; L < 5; ++L)
                for (int m1 = -l1; m1 <= l1; ++m1)
                    for (int m2 = -l2; m2 <= l2; ++m2)
                        t.v[l1][l2][L][m1 + 2][m2 + 2] = cgval(l1, m1, l2, m2, L, m1 + m2);
    return t;
}

constexpr int offl(int l) { return l * (4 * l * l - 1) / 3; }

// Sparse bilinear table: x2[K] += w * vn[i] * vn[j]  (i<=j, symmetric terms merged)
constexpr int MAXT = 3400;
struct Tab {
    int n;
    unsigned char ci[MAXT], cj[MAXT], ck[MAXT], np[MAXT];
    float w[MAXT];
};
constexpr Tab make_tab() {
    CgAll C = make_cg();
    Tab T{};
    T.n = 0;
    for (int i = 0; i < NIN; ++i) {
        int l1 = (i < 1) ? 0 : ((i < 10) ? 1 : 2);
        int r1 = i - offl(l1);
        int m = r1 / (2 * l1 + 1) - l1;
        int n = r1 % (2 * l1 + 1) - l1;
        for (int j = i; j < NIN; ++j) {
            int l2 = (j < 1) ? 0 : ((j < 10) ? 1 : 2);
            int r2 = j - offl(l2);
            int p = r2 / (2 * l2 + 1) - l2;
            int q = r2 % (2 * l2 + 1) - l2;
            bool first = true;
            for (int L = 0; L < 5; ++L) {
                int dl = l1 > l2 ? l1 - l2 : l2 - l1;
                if (L < dl || L > l1 + l2) continue;
                int M = m + p, N = n + q;
                if (M < -L || M > L || N < -L || N > L) continue;
                double w = C.v[l1][l2][L][m + 2][p + 2] * C.v[l1][l2][L][n + 2][q + 2];
                if (i != j)
                    w += C.v[l2][l1][L][p + 2][m + 2] * C.v[l2][l1][L][q + 2][n + 2];
                if (w < 1e-12 && w > -1e-12) continue;
                T.ci[T.n] = (unsigned char)i;
                T.cj[T.n] = (unsigned char)j;
                T.ck[T.n] = (unsigned char)(offl(L) + (M + L) * (2 * L + 1) + (N + L));
                T.np[T.n] = first ? 1 : 0;
                T.w[T.n]  = (float)w;
                first = false;
                ++T.n;
            }
        }
    }
    return T;
}
static constexpr Tab TAB = make_tab();
static constexpr int NNZ = TAB.n;
static_assert(NNZ > 0 && NNZ <= MAXT, "CG table size");

// ---------------- fully constant-folded FMA chain ---------------------------
// Every index is a template/constexpr constant so SROA keeps v[]/acc[] in VGPRs.
template <int T>
__device__ __forceinline__ void cg_step(const float (&v)[NIN], float (&acc)[NOUT], float& prod) {
    constexpr int   I  = TAB.ci[T];
    constexpr int   J  = TAB.cj[T];
    constexpr int   K  = TAB.ck[T];
    constexpr bool  NP = TAB.np[T] != 0;
    constexpr float W  = TAB.w[T];
    if constexpr (NP) prod = v[I] * v[J];
    acc[K] = fmaf(W, prod, acc[K]);
}

template <int... Ts>
__device__ __forceinline__ void cg_all(const float (&v)[NIN], float (&acc)[NOUT], float& prod,
                                       std::integer_sequence<int, Ts...>) {
    (cg_step<Ts>(v, acc, prod), ...);
}

// ---------------- CDNA5 async global->LDS path ------------------------------
typedef __attribute__((address_space(3))) float  lds_f32;
typedef __attribute__((address_space(1))) const float glb_f32;
typedef __attribute__((address_space(1))) int    g_i32;   // builtin wants non-const int*
typedef __attribute__((address_space(3))) int    l_i32;

#if defined(__has_builtin)
#if __has_builtin(__builtin_amdgcn_global_load_async_to_lds_b32)
#define HAVE_ASYNC_BUILTIN 1
#endif
#if __has_builtin(__builtin_amdgcn_s_wait_asynccnt)
#define HAVE_WAIT_BUILTIN 1
#endif
#endif

__device__ __forceinline__ void async_g2l_b32(glb_f32* g, lds_f32* l) {
#ifdef HAVE_ASYNC_BUILTIN
    __builtin_amdgcn_global_load_async_to_lds_b32((g_i32*)g, (l_i32*)l, 0, 0);
#else
    asm volatile("global_load_async_to_lds_b32 %0, %1, off"
                 :: "v"((unsigned)(uintptr_t)l), "v"(g)
                 : "memory");
#endif
}

__device__ __forceinline__ void wait_async0() {
#ifdef HAVE_WAIT_BUILTIN
    __builtin_amdgcn_s_wait_asynccnt(0);
#else
    asm volatile("s_wait_asynccnt 0" ::: "memory");
#endif
    asm volatile("" ::: "memory");   // compiler-level ordering fence
}

// ---------------- kernel ----------------------------------------------------
__global__ __launch_bounds__(THREADS)
void so3_local_act_kernel(const float* __restrict__ x,
                          const float* __restrict__ coeff,
                          float* __restrict__ out) {
    __shared__ float lds[NIN * THREADS];

    const int tid = threadIdx.x;
    const int blocksPerSlab = S_SPATIAL / THREADS;           // 54
    const int bf = blockIdx.x / blocksPerSlab;
    const int s  = (blockIdx.x % blocksPerSlab) * THREADS + tid;

    glb_f32* gin = (glb_f32*)(x + (size_t)bf * NIN * S_SPATIAL + s);
    float*   op  = out + (size_t)bf * NOUT * S_SPATIAL + s;

    // Stage all 35 channels through the async LDS pipe (ASYNCcnt), 128B/wave/channel.
    lds_f32* lbase = (lds_f32*)&lds[tid];
#pragma unroll
    for (int c = 0; c < NIN; ++c)
        async_g2l_b32(gin + (size_t)c * S_SPATIAL, lbase + c * THREADS);
    wait_async0();   // each lane only consumes its own slots -> no barrier needed

    // norm = sqrt(sum 8*pi^2 * x^2 / (2l+1) + eps) / 3
    const float W1 = 78.95683520871486f;          // 8*pi^2
    const float W3 = 78.95683520871486f / 3.0f;
    const float W5 = 78.95683520871486f / 5.0f;

    float v[NIN];
    float sum = 0.0f;
#pragma unroll
    for (int c = 0; c < NIN; ++c) {
        float t = lds[c * THREADS + tid];
        v[c] = t;
        float dw = (c < 1) ? W1 : ((c < 10) ? W3 : W5);
        sum += dw * t * t;
    }
    float r    = sqrtf(sum + 1e-12f);
    float norm = r * (1.0f / 3.0f);
    float inv  = 3.0f / r;
#pragma unroll
    for (int c = 0; c < NIN; ++c) v[c] *= inv;

    // Sparse CG bilinear form, fully constant-folded; acc[] and v[] live in VGPRs.
    float acc[NOUT];
#pragma unroll
    for (int k = 0; k < NOUT; ++k) acc[k] = 0.0f;
    float prod = 0.0f;
    cg_all(v, acc, prod, std::make_integer_sequence<int, NNZ>{});

    const float c0 = coeff[0] + 3.0f / 32.0f;
    const float c1 = coeff[1] + 0.5f;
    const float c2 = coeff[2] + 15.0f / 32.0f;

#pragma unroll
    for (int k = 0; k < NOUT; ++k) {
        float pv = c2 * acc[k];
        if (k < NIN) pv += c1 * v[k];
        if (k == 0)  pv += c0;
        op[(size_t)k * S_SPATIAL] = pv * norm;
    }
}

// ---------------- launch ----------------------------------------------------
extern "C" void kernel_launch(void* const* d_in, const int* in_sizes, int n_in,
                              void* d_out, int out_size, void* d_ws, size_t ws_size,
                              hipStream_t stream) {
    (void)in_sizes; (void)n_in; (void)out_size; (void)d_ws; (void)ws_size;
    const float* x     = (const float*)d_in[0];
    const float* coeff = (const float*)d_in[1];
    // d_in[2] = order (always 3; baked into the compile-time CG table)
    float* out = (float*)d_out;

    dim3 grid(NBF * (S_SPATIAL / THREADS));   // 32 * 54 = 1728 blocks
    so3_local_act_kernel<<<grid, THREADS, 0, stream>>>(x, coeff, out);
}